// Sparse_Spatial_Attention_73770358276426
// MI455X (gfx1250) — compile-verified
//
#include <hip/hip_runtime.h>
#include <math.h>

#define B_   2
#define T_   12
#define N_   512
#define F_   512
#define H_   8
#define DH_  64
#define KNB  16
#define SN_  27
#define BT_  (B_*T_)      /* 24   */
#define R_   (BT_*N_)     /* 12288 */
#define GRP  (H_*BT_)     /* 192  */
#define SCP  516          /* padded LDS row (floats) to avoid bank conflicts */

typedef __attribute__((ext_vector_type(2))) float v2f;
typedef __attribute__((ext_vector_type(8))) float v8f;

static __device__ __forceinline__ v8f wmma_f32_k4(v2f a, v2f b, v8f c) {
  // D = A(16x4,f32) * B(4x16,f32) + C(16x16,f32) ; exact fp32 matrix op on CDNA5
  return __builtin_amdgcn_wmma_f32_16x16x4_f32(false, a, false, b, (short)0, c,
                                               false, false);
}

// ---------------------------------------------------------------------------
// 1) x_ = x + spa_vec*spa_val + tem_vec*tem_val   (elementwise, n = row % N)
// ---------------------------------------------------------------------------
__global__ __launch_bounds__(256) void posenc_kernel(
    const float* __restrict__ x,
    const float* __restrict__ spaL, const float* __restrict__ spaV,
    const float* __restrict__ temL, const float* __restrict__ temV,
    float* __restrict__ xp) {
  int idx = blockIdx.x * 256 + threadIdx.x;        // over R_*F_
  int f = idx & (F_ - 1);
  int r = idx >> 9;                                 // F_ == 512
  int n = r & (N_ - 1);
  xp[idx] = x[idx] + spaV[n * F_ + f] * spaL[f] + temV[n * F_ + f] * temL[f];
}

// ---------------------------------------------------------------------------
// 2) Generic GEMM: out[r,o] = sum_i A[r,i]*W[o,i] + bias[o]  (+relu)(+resid)
//    Wave tile: 32(M) x 64(N) = 2 M-tiles x 4 N-tiles, 8 accumulators.
//    Each B fragment is reused by both M-tiles: 8 WMMAs per 6 b64 loads.
// ---------------------------------------------------------------------------
__global__ __launch_bounds__(256) void gemm512_kernel(
    const float* __restrict__ A, const float* __restrict__ W,
    const float* __restrict__ bias, const float* __restrict__ resid,
    float* __restrict__ out, int relu) {
  int waveId = blockIdx.x * 8 + (threadIdx.x >> 5);
  int tm = waveId >> 3;                 // 384 M tiles of 32 rows
  int tn = waveId & 7;                  // 8 N tiles of 64 cols
  int lane = threadIdx.x & 31;
  int m15 = lane & 15;                  // A: row-in-tile, B: col-in-tile
  int kh  = lane >> 4;                  // K half (0 -> K0,K1 ; 1 -> K2,K3)
  int row0 = tm * 32;
  int col0 = tn * 64;

  const float* arow0 = A + (size_t)(row0 + m15) * F_;
  const float* arow1 = arow0 + (size_t)16 * F_;
  v8f acc[2][4] = {};
  for (int kk = 0; kk < F_; kk += 4) {
    v2f a0 = *(const v2f*)(arow0 + kk + 2 * kh);
    v2f a1 = *(const v2f*)(arow1 + kk + 2 * kh);
#pragma unroll
    for (int j = 0; j < 4; ++j) {
      v2f b = *(const v2f*)(W + (size_t)(col0 + j * 16 + m15) * F_ + kk + 2 * kh);
      acc[0][j] = wmma_f32_k4(a0, b, acc[0][j]);
      acc[1][j] = wmma_f32_k4(a1, b, acc[1][j]);
    }
  }
#pragma unroll
  for (int mt = 0; mt < 2; ++mt) {
#pragma unroll
    for (int j = 0; j < 4; ++j) {
      int nc = col0 + j * 16 + m15;
      float bvv = bias[nc];
#pragma unroll
      for (int i = 0; i < 8; ++i) {
        int r = row0 + mt * 16 + i + 8 * kh;   // C layout: VGPR i -> M=i+8*half
        float v = acc[mt][j][i] + bvv;
        if (relu) v = fmaxf(v, 0.0f);
        if (resid) v += resid[(size_t)r * F_ + nc];
        out[(size_t)r * F_ + nc] = v;
      }
    }
  }
}

// ---------------------------------------------------------------------------
// 3) qks + M: M[h,bt,n] = sum_k (Q[n] . K[adj[n,k]]) * w_proj[k] + b_proj
// ---------------------------------------------------------------------------
__global__ __launch_bounds__(256) void qks_kernel(
    const float* __restrict__ Q, const float* __restrict__ K,
    const int* __restrict__ adj, const float* __restrict__ wproj,
    const float* __restrict__ bproj, float* __restrict__ Mout) {
  int g = blockIdx.x * 256 + threadIdx.x;          // H_*BT_*N_
  int n  = g & (N_ - 1);
  int bt = (g >> 9) % BT_;
  int h  = g / (BT_ * N_);
  const float* q = Q + ((size_t)(bt * N_ + n)) * F_ + h * DH_;
  float acc = bproj[0];
  for (int k = 0; k < KNB; ++k) {
    int nb = adj[n * KNB + k];
    const float* kr = K + ((size_t)(bt * N_ + nb)) * F_ + h * DH_;
    float dot = 0.0f;
#pragma unroll 8
    for (int d = 0; d < DH_; ++d) dot += q[d] * kr[d];
    acc += dot * wproj[k];
  }
  Mout[g] = acc;
}

// ---------------------------------------------------------------------------
// 4) top-SN per (h,bt): selection, strict '>' => lowest index on ties
// ---------------------------------------------------------------------------
__global__ void topk_kernel(const float* __restrict__ M, int* __restrict__ top) {
  int g = blockIdx.x * blockDim.x + threadIdx.x;
  if (g >= GRP) return;
  const float* m = M + (size_t)g * N_;
  unsigned used[16] = {0};
  for (int s = 0; s < SN_; ++s) {
    float best = -3.4e38f; int bi = 0;
    for (int n = 0; n < N_; ++n) {
      if ((used[n >> 5] >> (n & 31)) & 1u) continue;
      float v = m[n];
      if (v > best) { best = v; bi = n; }
    }
    used[bi >> 5] |= 1u << (bi & 31);
    top[g * SN_ + s] = bi;
  }
}

// ---------------------------------------------------------------------------
// 5) Attention core per (h,bt): scores(WMMA) -> softmax -> argmax -> val(WMMA)
// ---------------------------------------------------------------------------
__global__ __launch_bounds__(256) void attn_kernel(
    const float* __restrict__ Q, const float* __restrict__ Km,
    const float* __restrict__ Vm, const int* __restrict__ top,
    int* __restrict__ cp, float* __restrict__ valout) {
  __shared__ float sc[32][SCP];                    // 32 x 512 scores (padded)
  int g   = blockIdx.x;                            // h*BT_ + bt
  int bt  = g % BT_;
  int h   = g / BT_;
  int tid = threadIdx.x;
  int w   = tid >> 5;                              // wave 0..7
  int lane = tid & 31;
  int m15 = lane & 15;
  int kh  = lane >> 4;
  const int* tp = top + g * SN_;

  // ---- scores = Qr * Kh^T / sqrt(Dh): M=32(27 live) N=512 K=64 -------------
  for (int mt = 0; mt < 2; ++mt) {
    int srow = mt * 16 + m15;
    int qidx = (srow < SN_) ? tp[srow] : -1;       // rows >=27 zero-padded
    const float* qrow = Q + ((size_t)(bt * N_ + (qidx >= 0 ? qidx : 0))) * F_ + h * DH_;
    for (int nt = w; nt < 32; nt += 8) {
      int n0 = nt * 16;
      v8f acc = {};
      for (int kk = 0; kk < DH_; kk += 4) {
        v2f a;
        if (qidx >= 0) a = *(const v2f*)(qrow + kk + 2 * kh);
        else { a.x = 0.0f; a.y = 0.0f; }
        v2f b = *(const v2f*)(Km + ((size_t)(bt * N_ + n0 + m15)) * F_ +
                              h * DH_ + kk + 2 * kh);
        acc = wmma_f32_k4(a, b, acc);
      }
#pragma unroll
      for (int i = 0; i < 8; ++i) {
        int r = mt * 16 + i + 8 * kh;
        sc[r][n0 + m15] = acc[i] * 0.125f;         // 1/sqrt(64)
      }
    }
  }
  __syncthreads();

  // ---- softmax over N per live row (one wave per row) ----------------------
  for (int s = w; s < SN_; s += 8) {
    float mx = -3.4e38f;
#pragma unroll
    for (int i = 0; i < 16; ++i) mx = fmaxf(mx, sc[s][lane + 32 * i]);
    for (int off = 16; off; off >>= 1) mx = fmaxf(mx, __shfl_xor(mx, off, 32));
    float sum = 0.0f;
#pragma unroll
    for (int i = 0; i < 16; ++i) {
      float e = __expf(sc[s][lane + 32 * i] - mx);
      sc[s][lane + 32 * i] = e;
      sum += e;
    }
    for (int off = 16; off; off >>= 1) sum += __shfl_xor(sum, off, 32);
    float inv = 1.0f / sum;
#pragma unroll
    for (int i = 0; i < 16; ++i) sc[s][lane + 32 * i] *= inv;
  }
  __syncthreads();

  // ---- cp[n] = argmax_s attn[s][n]  (normalized values; first-index ties) --
  for (int n = tid; n < N_; n += 256) {
    float best = sc[0][n]; int bi = 0;
    for (int s = 1; s < SN_; ++s) {
      float v = sc[s][n];
      if (v > best) { best = v; bi = s; }
    }
    cp[g * N_ + n] = bi;
  }

  // ---- val = attn * Vh : M=32 N=64 K=512, one 16x16 tile per wave ----------
  {
    int mt = w >> 2;
    int d0 = (w & 3) * 16;
    const float* arow = &sc[mt * 16 + m15][0];
    v8f acc = {};
    for (int kk = 0; kk < N_; kk += 4) {
      v2f a = *(const v2f*)(arow + kk + 2 * kh);
      size_t vb = ((size_t)(bt * N_ + kk + 2 * kh)) * F_ + h * DH_ + d0 + m15;
      v2f b;
      b.x = Vm[vb];
      b.y = Vm[vb + F_];
      acc = wmma_f32_k4(a, b, acc);
    }
#pragma unroll
    for (int i = 0; i < 8; ++i) {
      int r = mt * 16 + i + 8 * kh;
      valout[((size_t)g * 32 + r) * DH_ + d0 + m15] = acc[i];
    }
  }
}

// ---------------------------------------------------------------------------
// 6) Gather: out[b,t,n, h*Dh+d] = val[h,bt, cp[h,bt,n], d]
// ---------------------------------------------------------------------------
__global__ __launch_bounds__(256) void gather_kernel(
    const float* __restrict__ val, const int* __restrict__ cp,
    float* __restrict__ out) {
  int idx = blockIdx.x * 256 + threadIdx.x;        // over R_*F_
  int f = idx & (F_ - 1);
  int r = idx >> 9;
  int h = f >> 6;
  int d = f & 63;
  int n = r & (N_ - 1);
  int bt = r >> 9;
  int g = h * BT_ + bt;
  int s = cp[g * N_ + n];
  out[idx] = val[((size_t)g * 32 + s) * DH_ + d];
}

// ---------------------------------------------------------------------------
// 7) LayerNorm per row (one wave per row); gamma==nullptr -> no affine
// ---------------------------------------------------------------------------
__global__ __launch_bounds__(256) void ln_kernel(
    const float* __restrict__ in, const float* __restrict__ gamma,
    const float* __restrict__ beta, float* __restrict__ out) {
  int row = blockIdx.x * 8 + (threadIdx.x >> 5);
  int lane = threadIdx.x & 31;
  const float* p = in + (size_t)row * F_;
  float s = 0.0f, s2 = 0.0f;
#pragma unroll
  for (int i = 0; i < 16; ++i) {
    float v = p[lane + 32 * i];
    s += v; s2 += v * v;
  }
  for (int off = 16; off; off >>= 1) {
    s  += __shfl_xor(s, off, 32);
    s2 += __shfl_xor(s2, off, 32);
  }
  float mu  = s * (1.0f / F_);
  float var = s2 * (1.0f / F_) - mu * mu;
  float rstd = rsqrtf(var + 1e-5f);
  float* o = out + (size_t)row * F_;
#pragma unroll
  for (int i = 0; i < 16; ++i) {
    int f = lane + 32 * i;
    float y = (p[f] - mu) * rstd;
    if (gamma) y = y * gamma[f] + beta[f];
    o[f] = y;
  }
}

// ---------------------------------------------------------------------------
extern "C" void kernel_launch(void* const* d_in, const int* in_sizes, int n_in,
                              void* d_out, int out_size, void* d_ws, size_t ws_size,
                              hipStream_t stream) {
  (void)in_sizes; (void)n_in; (void)out_size; (void)ws_size;
  const float* x       = (const float*)d_in[0];
  const float* spa_val = (const float*)d_in[1];
  const float* spa_vec = (const float*)d_in[2];
  const float* tem_val = (const float*)d_in[3];
  const float* tem_vec = (const float*)d_in[4];
  const int*   adj     = (const int*)d_in[5];
  const float* Wq = (const float*)d_in[6];  const float* bq = (const float*)d_in[7];
  const float* Wk = (const float*)d_in[8];  const float* bk = (const float*)d_in[9];
  const float* Wv = (const float*)d_in[10]; const float* bv = (const float*)d_in[11];
  const float* Wo = (const float*)d_in[12]; const float* bo = (const float*)d_in[13];
  const float* wproj = (const float*)d_in[14];
  const float* bproj = (const float*)d_in[15];
  const float* gamma = (const float*)d_in[16];
  const float* beta  = (const float*)d_in[17];
  const float* W1 = (const float*)d_in[18]; const float* b1 = (const float*)d_in[19];
  const float* W2 = (const float*)d_in[20]; const float* b2 = (const float*)d_in[21];
  float* out = (float*)d_out;

  const size_t RF = (size_t)R_ * F_;
  float* ws  = (float*)d_ws;
  float* XP  = ws;             // x_ ; reused: attn_out, then h1
  float* Qb  = ws + RF;
  float* Kb  = ws + 2 * RF;    // reused: GEMM tmp after attention
  float* Vb  = ws + 3 * RF;    // reused: v (post-LN) after attention
  float* Mb  = ws + 4 * RF;                         // GRP*N_ floats
  int*   TOP = (int*)(Mb + (size_t)GRP * N_);       // GRP*SN_ ints
  int*   CP  = TOP + GRP * SN_;                     // GRP*N_ ints
  float* VAL = (float*)(CP + (size_t)GRP * N_);     // GRP*32*DH_ floats

  dim3 blk(256);
  const int gemmBlocks = R_ / 32;                   // 384: 32-row wave tiles
  // x_ = x + positional encodings
  posenc_kernel<<<RF / 256, blk, 0, stream>>>(x, spa_val, spa_vec, tem_val, tem_vec, XP);
  // Q, K, V projections (fp32 WMMA GEMMs)
  gemm512_kernel<<<gemmBlocks, blk, 0, stream>>>(XP, Wq, bq, nullptr, Qb, 0);
  gemm512_kernel<<<gemmBlocks, blk, 0, stream>>>(XP, Wk, bk, nullptr, Kb, 0);
  gemm512_kernel<<<gemmBlocks, blk, 0, stream>>>(XP, Wv, bv, nullptr, Vb, 0);
  // sparse neighborhood scores + top-k row selection
  qks_kernel<<<(GRP * N_) / 256, blk, 0, stream>>>(Qb, Kb, adj, wproj, bproj, Mb);
  topk_kernel<<<1, GRP, 0, stream>>>(Mb, TOP);
  // attention core (scores WMMA / softmax / argmax / val WMMA)
  attn_kernel<<<GRP, blk, 0, stream>>>(Qb, Kb, Vb, TOP, CP, VAL);
  // gather selected rows back to (B,T,N,F)  -> XP
  gather_kernel<<<RF / 256, blk, 0, stream>>>(VAL, CP, XP);
  // output projection + LN(gamma,beta)
  gemm512_kernel<<<gemmBlocks, blk, 0, stream>>>(XP, Wo, bo, nullptr, Kb, 0);
  ln_kernel<<<R_ / 8, blk, 0, stream>>>(Kb, gamma, beta, Vb);      // Vb = v
  // FFN: h1 = relu(v*W1^T + b1) ; h2 = h1*W2^T + b2 + v ; out = LN(h2)
  gemm512_kernel<<<gemmBlocks, blk, 0, stream>>>(Vb, W1, b1, nullptr, XP, 1);
  gemm512_kernel<<<gemmBlocks, blk, 0, stream>>>(XP, W2, b2, Vb, Kb, 0);
  ln_kernel<<<R_ / 8, blk, 0, stream>>>(Kb, nullptr, nullptr, out);
}